// GAEOneHop_51728586113709
// MI455X (gfx1250) — compile-verified
//
#include <hip/hip_runtime.h>

typedef float v2f __attribute__((ext_vector_type(2)));
typedef float v8f __attribute__((ext_vector_type(8)));

#define H128 128

static inline int cdiv_l(long a, long b) { return (int)((a + b - 1) / b); }

// Hardware float atomic add (global_atomic_add_f32) — avoid CAS-loop fallback.
__device__ __forceinline__ void atomic_add_f32(float* p, float v) {
#if defined(__HIP_DEVICE_COMPILE__)
    unsafeAtomicAdd(p, v);
#else
    atomicAdd(p, v);
#endif
}

// ---------------------------------------------------------------------------
// Elementwise helpers
// ---------------------------------------------------------------------------
__global__ __launch_bounds__(256) void fill_f32(float* __restrict__ p, float v, long n) {
    long i = (long)blockIdx.x * 256 + threadIdx.x;
    if (i < n) p[i] = v;
}

__global__ __launch_bounds__(256) void count_idx(const int* __restrict__ idx,
                                                 float* __restrict__ deg, int n) {
    int i = blockIdx.x * 256 + threadIdx.x;
    if (i < n) atomic_add_f32(&deg[idx[i]], 1.0f);
}

__global__ __launch_bounds__(256) void rsqrt_f32(const float* __restrict__ a,
                                                 float* __restrict__ o, int n) {
    int i = blockIdx.x * 256 + threadIdx.x;
    if (i < n) o[i] = rsqrtf(a[i]);
}

// out[i,:] = in[perm[i],:]
__global__ __launch_bounds__(256) void gather_rows(const float* __restrict__ in,
                                                   const int* __restrict__ perm,
                                                   float* __restrict__ out, int M) {
    long i = (long)blockIdx.x * 256 + threadIdx.x;
    if (i >= (long)M * H128) return;
    int r = (int)(i >> 7), c = (int)(i & 127);
    out[i] = in[(size_t)perm[r] * H128 + c];
}

// out[perm[i],:] = in[i,:]   (out pre-zeroed)
__global__ __launch_bounds__(256) void scatter_rows(const float* __restrict__ in,
                                                    const int* __restrict__ perm,
                                                    float* __restrict__ out, int M) {
    long i = (long)blockIdx.x * 256 + threadIdx.x;
    if (i >= (long)M * H128) return;
    int r = (int)(i >> 7), c = (int)(i & 127);
    out[(size_t)perm[r] * H128 + c] = in[i];
}

// ---------------------------------------------------------------------------
// GCN edge aggregation: agg[col] += dinv[row]*dinv[col] * t[row]
// One wave32 per edge; each lane moves a float4 (4 x hardware f32 atomic add).
// ---------------------------------------------------------------------------
__global__ __launch_bounds__(256) void scatter_edges(const int* __restrict__ rows,
                                                     const int* __restrict__ cols,
                                                     const float* __restrict__ dinv,
                                                     const float* __restrict__ t,
                                                     float* __restrict__ agg, int E) {
    int e = blockIdx.x * 8 + (int)(threadIdx.x >> 5);
    if (e >= E) return;
    int lane = threadIdx.x & 31;
    int r = rows[e];
    int c = cols[e];
    float norm = dinv[r] * dinv[c];
    float4 v = *((const float4*)(t + (size_t)r * H128) + lane);
    float* dst = agg + (size_t)c * H128 + lane * 4;
    atomic_add_f32(dst + 0, norm * v.x);
    atomic_add_f32(dst + 1, norm * v.y);
    atomic_add_f32(dst + 2, norm * v.z);
    atomic_add_f32(dst + 3, norm * v.w);
}

// out = maybe_relu(agg + 2*dinv^2 * t + bias)
__global__ __launch_bounds__(256) void combine_k(const float* __restrict__ agg,
                                                 const float* __restrict__ t,
                                                 const float* __restrict__ dinv,
                                                 const float* __restrict__ bias,
                                                 float* __restrict__ out, long total,
                                                 int relu) {
    long i = (long)blockIdx.x * 256 + threadIdx.x;
    if (i >= total) return;
    int n = (int)(i >> 7), c = (int)(i & 127);
    float d = dinv[n];
    float v = agg[i] + 2.0f * d * d * t[i] + bias[c];
    out[i] = relu ? fmaxf(v, 0.0f) : v;
}

// ---------------------------------------------------------------------------
// FP32 WMMA GEMM:  C[N,128] = A[N,K] @ W[K,128]  (+ deg[n]*wextra[col]) (+ bias[col])
// One wave computes a 16x128 strip: 8 accumulator tiles, K-loop step 4 using
// V_WMMA_F32_16X16X4_F32.
//   A frag (16x4):  lane l: g=l>>4, m=l&15;  a = {A[m][k0+2g], A[m][k0+2g+1]}
//   B frag (4x16):  lane l: g=l>>4, n=l&15;  b = {W[k0+2g][n], W[k0+2g+1][n]}
//   C tile (16x16): vgpr r, lane l: C[r+8g][n]
// ---------------------------------------------------------------------------
__global__ __launch_bounds__(256) void gemm_wmma_f32(const float* __restrict__ A,
                                                     const float* __restrict__ W,
                                                     const float* __restrict__ wextra,
                                                     const float* __restrict__ deg,
                                                     const float* __restrict__ bias,
                                                     float* __restrict__ C, int N, int K) {
    const int lane = threadIdx.x & 31;
    const int wave = threadIdx.x >> 5;
    const int rowBase = (blockIdx.x * 8 + wave) * 16;
    if (rowBase >= N) return;  // wave-uniform: EXEC stays all-ones for WMMA
    const int g = lane >> 4;
    const int ln = lane & 15;

    const v8f vz = {0.f, 0.f, 0.f, 0.f, 0.f, 0.f, 0.f, 0.f};
    v8f acc[8];
#pragma unroll
    for (int t = 0; t < 8; ++t) acc[t] = vz;

    const float* arow = A + (size_t)(rowBase + ln) * K;
    for (int k0 = 0; k0 < K; k0 += 4) {
        v2f a = *(const v2f*)(arow + k0 + 2 * g);
        const float* wp = W + (size_t)(k0 + 2 * g) * H128 + ln;
#pragma unroll
        for (int t = 0; t < 8; ++t) {
            v2f b;
            b.x = wp[t * 16];
            b.y = wp[t * 16 + H128];
            acc[t] = __builtin_amdgcn_wmma_f32_16x16x4_f32(
                false, a, false, b, (short)0, acc[t], false, false);
        }
    }

    float degv[8];
    if (deg != nullptr) {
#pragma unroll
        for (int r = 0; r < 8; ++r) degv[r] = deg[rowBase + r + 8 * g];
    }

#pragma unroll
    for (int t = 0; t < 8; ++t) {
        int col = t * 16 + ln;
        float wex = (deg != nullptr) ? wextra[col] : 0.0f;
        float bv = (bias != nullptr) ? bias[col] : 0.0f;
#pragma unroll
        for (int r = 0; r < 8; ++r) {
            float v = acc[t][r];
            if (deg != nullptr) v += degv[r] * wex;
            v += bv;
            C[(size_t)(rowBase + r + 8 * g) * H128 + col] = v;
        }
    }
}

// ---------------------------------------------------------------------------
// Host-side orchestration
// ---------------------------------------------------------------------------
extern "C" void kernel_launch(void* const* d_in, const int* in_sizes, int n_in,
                              void* d_out, int out_size, void* d_ws, size_t ws_size,
                              hipStream_t stream) {
    const int N0 = 80000, N1 = 60000, N2 = 45000;
    const int E0 = 1280000, E1 = 360000, E2 = 315000;
    const int CIN = 64, H = 128;
    (void)in_sizes; (void)n_in; (void)out_size; (void)ws_size;

    const float* x    = (const float*)d_in[0];
    const float* Wd0  = (const float*)d_in[2];
    const float* bd0  = (const float*)d_in[3];
    const float* Wd1  = (const float*)d_in[4];
    const float* bd1  = (const float*)d_in[5];
    const float* Wu0  = (const float*)d_in[6];
    const float* bu0  = (const float*)d_in[7];
    const float* Wu1  = (const float*)d_in[8];
    const float* bu1  = (const float*)d_in[9];
    const float* Wlin = (const float*)d_in[10];
    const float* blin = (const float*)d_in[11];
    const int* ei0   = (const int*)d_in[12];
    const int* ei1   = (const int*)d_in[13];
    const int* ei2   = (const int*)d_in[14];
    const int* perm1 = (const int*)d_in[15];
    const int* perm2 = (const int*)d_in[16];

    float* bufA = (float*)d_ws;
    float* bufB = bufA + (size_t)N0 * H;
    float* bufC = bufB + (size_t)N0 * H;
    float* dinv = bufC + (size_t)N0 * H;
    float* degt = dinv + N0;
    float* degdeep = degt + N0;

    auto run_dinv = [&](const int* colptr, int E, int N) {
        fill_f32<<<cdiv_l(N, 256), 256, 0, stream>>>(degt, 2.0f, N);
        count_idx<<<cdiv_l(E, 256), 256, 0, stream>>>(colptr, degt, E);
        rsqrt_f32<<<cdiv_l(N, 256), 256, 0, stream>>>(degt, dinv, N);
    };
    auto run_gemm = [&](const float* A, const float* W, const float* wex,
                        const float* dg, const float* bias, float* Cp, int N, int K) {
        gemm_wmma_f32<<<cdiv_l(N, 128), 256, 0, stream>>>(A, W, wex, dg, bias, Cp, N, K);
    };
    auto run_gcn_rest = [&](const int* ei, int E, int N, float* t, float* agg,
                            const float* bias) {
        // t = h@W already in `t`; dinv already computed
        fill_f32<<<cdiv_l((long)N * H, 256), 256, 0, stream>>>(agg, 0.0f, (long)N * H);
        scatter_edges<<<cdiv_l(E, 8), 256, 0, stream>>>(ei, ei + E, dinv, t, agg, E);
        combine_k<<<cdiv_l((long)N * H, 256), 256, 0, stream>>>(agg, t, dinv, bias, agg,
                                                                (long)N * H, 1);
    };

    // degree from deepest edge index (reused at both up levels — faithful bug)
    fill_f32<<<cdiv_l(N0, 256), 256, 0, stream>>>(degdeep, 0.0f, N0);
    count_idx<<<cdiv_l(2 * E2, 256), 256, 0, stream>>>(ei2, degdeep, 2 * E2);

    // ---- down level 0: bufB = relu(gcn(x, ei0, Wd0, bd0)) ----
    run_gemm(x, Wd0, nullptr, nullptr, nullptr, bufA, N0, CIN);
    run_dinv(ei0 + E0, E0, N0);
    run_gcn_rest(ei0, E0, N0, bufA, bufB, bd0);

    // pool: bufA = bufB[perm1]  (N1 rows)
    gather_rows<<<cdiv_l((long)N1 * H, 256), 256, 0, stream>>>(bufB, perm1, bufA, N1);

    // ---- down level 1: bufB = relu(gcn(bufA, ei1, Wd1, bd1)) ----
    run_gemm(bufA, Wd1, nullptr, nullptr, nullptr, bufC, N1, H);
    run_dinv(ei1 + E1, E1, N1);
    run_gcn_rest(ei1, E1, N1, bufC, bufB, bd1);

    // pool: bufA = bufB[perm2]  (N2 rows)
    gather_rows<<<cdiv_l((long)N2 * H, 256), 256, 0, stream>>>(bufB, perm2, bufA, N2);

    // ---- up level 1: unpool + [up, deg] @ Wu0, gcn over ei1 ----
    fill_f32<<<cdiv_l((long)N1 * H, 256), 256, 0, stream>>>(bufC, 0.0f, (long)N1 * H);
    scatter_rows<<<cdiv_l((long)N2 * H, 256), 256, 0, stream>>>(bufA, perm2, bufC, N2);
    run_gemm(bufC, Wu0, Wu0 + (size_t)H * H, degdeep, nullptr, bufA, N1, H);
    // dinv for ei1 unchanged since down level 1 — reuse (identical inputs).
    run_gcn_rest(ei1, E1, N1, bufA, bufB, bu0);

    // ---- up level 0: unpool + [up, deg] @ Wu1, gcn over ei0 ----
    fill_f32<<<cdiv_l((long)N0 * H, 256), 256, 0, stream>>>(bufC, 0.0f, (long)N0 * H);
    scatter_rows<<<cdiv_l((long)N1 * H, 256), 256, 0, stream>>>(bufB, perm1, bufC, N1);
    run_gemm(bufC, Wu1, Wu1 + (size_t)H * H, degdeep, nullptr, bufA, N0, H);
    run_dinv(ei0 + E0, E0, N0);  // dinv currently holds ei1's values — recompute
    run_gcn_rest(ei0, E0, N0, bufA, bufB, bu1);

    // ---- final linear into d_out ----
    run_gemm(bufB, Wlin, nullptr, nullptr, blin, (float*)d_out, N0, H);
}